// Temp2SpaceNN_48198122996182
// MI455X (gfx1250) — compile-verified
//
#include <hip/hip_runtime.h>
#include <hip/hip_bf16.h>
#include <cstdint>

typedef __attribute__((ext_vector_type(2))) float v2f;
typedef __attribute__((ext_vector_type(8))) float v8f;

#define BN_EPS 1e-5f

// Async global->LDS copy (gfx1250). vdst = workgroup-relative LDS byte offset
// (low 32 bits of a generic pointer to __shared__), GV addressing mode.
__device__ __forceinline__ void async_copy_b128(uint32_t lds_off, const void* gaddr) {
    asm volatile("global_load_async_to_lds_b128 %0, %1, off"
                 :: "v"(lds_off), "v"((uint64_t)(uintptr_t)gaddr) : "memory");
}
__device__ __forceinline__ void async_wait0() {
    asm volatile("s_wait_asynccnt 0x0" ::: "memory");
}
__device__ __forceinline__ uint32_t lds_off_u32(const void* p) {
    return (uint32_t)(uintptr_t)p;   // low 32 bits of flat shared ptr = LDS offset
}

// ---------------------------------------------------------------------------
// K0: zero the BN statistic accumulators (must be done every call)
// ---------------------------------------------------------------------------
__global__ void zero_stats_kernel(float* __restrict__ stats) {
    int t = threadIdx.x;
    if (t < 192) stats[t] = 0.0f;
}

// ---------------------------------------------------------------------------
// K1: conv1 (8 -> 32 ch, K=5, pad=2), direct, + per-channel sum/sumsq
// grid: 16 n * 4 ltiles = 64 blocks, 256 threads (one l per thread)
// h1 layout: (n, co, l)
// ---------------------------------------------------------------------------
__global__ void conv1_kernel(const float* __restrict__ x, const float* __restrict__ w,
                             const float* __restrict__ bias, float* __restrict__ h1,
                             float* __restrict__ ssum, float* __restrict__ ssq) {
    __shared__ float wsh[32 * 8 * 5];
    __shared__ float ls_sum[32], ls_sq[32];
    int t = threadIdx.x;
    for (int i = t; i < 1280; i += 256) wsh[i] = w[i];
    if (t < 32) { ls_sum[t] = 0.0f; ls_sq[t] = 0.0f; }
    __syncthreads();

    int n = blockIdx.x >> 2;
    int l = ((blockIdx.x & 3) << 8) + t;

    float xv[5][8];
    for (int tap = 0; tap < 5; ++tap) {
        int ll = l + tap - 2;
        bool ok = (ll >= 0) && (ll < 1024);
        for (int ci = 0; ci < 8; ++ci)
            xv[tap][ci] = ok ? x[((n << 10) + ll) * 8 + ci] : 0.0f;
    }
    for (int co = 0; co < 32; ++co) {
        float acc = bias[co];
        const float* wr = &wsh[co * 40];
        for (int ci = 0; ci < 8; ++ci)
            for (int tap = 0; tap < 5; ++tap)
                acc = fmaf(xv[tap][ci], wr[ci * 5 + tap], acc);
        h1[(((n << 5) + co) << 10) + l] = acc;
        atomicAdd(&ls_sum[co], acc);
        atomicAdd(&ls_sq[co], acc * acc);
    }
    __syncthreads();
    if (t < 32) { atomicAdd(&ssum[t], ls_sum[t]); atomicAdd(&ssq[t], ls_sq[t]); }
}

// ---------------------------------------------------------------------------
// K2: fold batch stats into per-channel scale/shift: y = x*scale + shift
// ---------------------------------------------------------------------------
__global__ void bn_finalize_kernel(int count, float invN,
                                   const float* __restrict__ sum, const float* __restrict__ sq,
                                   const float* __restrict__ g, const float* __restrict__ b,
                                   float* __restrict__ scale, float* __restrict__ shift) {
    int t = threadIdx.x + blockIdx.x * blockDim.x;
    if (t < count) {
        float m = sum[t] * invN;
        float v = sq[t] * invN - m * m;
        float s = g[t] * rsqrtf(v + BN_EPS);
        scale[t] = s;
        shift[t] = b[t] - m * s;
    }
}

// ---------------------------------------------------------------------------
// K3: conv2 as implicit GEMM on WMMA: D(64co x 16384) = W(64x160) x im2col(160 x 16384)
// BN1 scale/shift + ReLU fused into the LDS staging of h1.
// grid: 16 n * 32 ltiles(32) = 512 blocks, 256 threads = 8 waves.
// ---------------------------------------------------------------------------
__global__ void conv2_wmma_kernel(const float* __restrict__ h1, const float* __restrict__ w,
                                  const float* __restrict__ bias,
                                  const float* __restrict__ sc1, const float* __restrict__ sh1,
                                  float* __restrict__ h2,
                                  float* __restrict__ ssum, float* __restrict__ ssq) {
    __shared__ float wsh[64 * 160];   // 40 KB weights, k = ci*5+tap matches (64,32,5) flat
    __shared__ float ins[32][40];     // bn1+relu'd h1 slice: ci x u, u in [0,36)
    __shared__ float ls_sum[64], ls_sq[64];
    int t = threadIdx.x;
    int n  = blockIdx.x >> 5;
    int lb = (blockIdx.x & 31) << 5;  // l tile base (32 wide)

    for (int i = t; i < 10240; i += 256) wsh[i] = w[i];
    if (t < 64) { ls_sum[t] = 0.0f; ls_sq[t] = 0.0f; }
    for (int i = t; i < 32 * 36; i += 256) {
        int ci = i / 36, u = i - ci * 36;
        int l = lb - 2 + u;
        float v = 0.0f;
        if (l >= 0 && l < 1024) v = h1[(((n << 5) + ci) << 10) + l];
        v = fmaf(v, sc1[ci], sh1[ci]);    // BN1
        ins[ci][u] = v > 0.0f ? v : 0.0f; // ReLU
    }
    __syncthreads();

    int wave = t >> 5, lane = t & 31;
    int cobase = (wave >> 1) << 4;    // 0,16,32,48
    int sub = (wave & 1) << 4;        // l sub-tile: 0 or 16
    int kg = lane >> 4, mn = lane & 15;

    v8f c = {};
    for (int ks = 0; ks < 40; ++ks) {
        int k0 = (ks << 2) + (kg << 1);
        int k1 = k0 + 1;
        v2f a, bm;
        a.x = wsh[(cobase + mn) * 160 + k0];
        a.y = wsh[(cobase + mn) * 160 + k1];
        int ci0 = k0 / 5, t0 = k0 - ci0 * 5;
        int ci1 = k1 / 5, t1 = k1 - ci1 * 5;
        bm.x = ins[ci0][sub + mn + t0];
        bm.y = ins[ci1][sub + mn + t1];
        c = __builtin_amdgcn_wmma_f32_16x16x4_f32(false, a, false, bm, (short)0, c, false, false);
    }

    for (int r = 0; r < 8; ++r) {
        int co = cobase + r + (kg << 3);
        int l  = lb + sub + mn;
        float v = c[r] + bias[co];
        h2[(((n << 6) + co) << 10) + l] = v;
        atomicAdd(&ls_sum[co], v);
        atomicAdd(&ls_sq[co], v * v);
    }
    __syncthreads();
    if (t < 64) { atomicAdd(&ssum[t], ls_sum[t]); atomicAdd(&ssq[t], ls_sq[t]); }
}

// ---------------------------------------------------------------------------
// K4: max-product pooling z[n,j,k] = max_i b[n,i,j] * relu(bn2(h2[n,k,i]))
// (max,*) semiring -> VALU with LDS tiling. grid: 16 n * 4 jtiles = 64 blocks.
// b tile staged with GLOBAL_LOAD_ASYNC_TO_LDS_B128 (raw copy, 16B-aligned rows);
// h tile staged through VALU (BN2+ReLU fused into the store).
// ---------------------------------------------------------------------------
__global__ void pool_kernel(const float* __restrict__ bmat, const float* __restrict__ h2,
                            const float* __restrict__ sc2, const float* __restrict__ sh2,
                            float* __restrict__ z) {
    __shared__ __align__(16) float hs[64][65];   // [i][k]
    __shared__ __align__(16) float bs[64][32];   // [i][j] raw copy target (row = 128B)
    int t = threadIdx.x;
    int n  = blockIdx.x >> 2;
    int jb = (blockIdx.x & 3) << 5;
    int jj = t & 31;
    int kgrp = t >> 5;

    float acc[8];
    for (int q = 0; q < 8; ++q) acc[q] = 0.0f;  // all products >= 0 (b>=0, relu>=0)

    for (int ic = 0; ic < 16; ++ic) {
        int ibase = ic << 6;
        // async stage b chunk: 64 rows x 32 floats = 512 x b128
        for (int i = 0; i < 2; ++i) {
            int idx = (i << 8) + t;          // 0..511
            int ii = idx >> 3;
            int q  = (idx & 7) << 2;         // float offset within row
            const float* g = bmat + (size_t)(((n << 10) + ibase + ii) << 7) + jb + q;
            async_copy_b128(lds_off_u32(&bs[ii][q]), g);
        }
        // VALU stage h2 chunk with BN2+ReLU fused (coalesced over i)
        for (int idx = t; idx < 4096; idx += 256) {
            int k = idx >> 6, ii = idx & 63;
            float v = h2[(((n << 6) + k) << 10) + ibase + ii];
            v = fmaf(v, sc2[k], sh2[k]);
            hs[ii][k] = v > 0.0f ? v : 0.0f;
        }
        async_wait0();
        __syncthreads();
        for (int ii = 0; ii < 64; ++ii) {
            float bv = bs[ii][jj];
            for (int q = 0; q < 8; ++q) {
                float p = bv * hs[ii][(kgrp << 3) + q];
                acc[q] = p > acc[q] ? p : acc[q];
            }
        }
        __syncthreads();
    }
    for (int q = 0; q < 8; ++q)
        z[(n << 13) + (jb + jj) * 64 + (kgrp << 3) + q] = acc[q];
}

// ---------------------------------------------------------------------------
// K5: fc1 with WMMA: zf(16x100) = z(16x8192) @ w1^T + b1
// grid: 7 col-tiles of 16, 256 threads = 8 waves splitting K.
// 256-wide K chunks staged with async-to-LDS b128 (rows padded to 260 floats =
// 1040 B: 16B-aligned transfers, stride 4 banks -> conflict-free reads).
// Cross-wave reduce through LDS at the end.
// ---------------------------------------------------------------------------
__global__ void fc1_wmma_kernel(const float* __restrict__ z, const float* __restrict__ w1,
                                const float* __restrict__ b1, float* __restrict__ zf) {
    __shared__ __align__(16) float zs[16][260];
    __shared__ __align__(16) float w1s[16][260];
    __shared__ float red[8][256];
    int t = threadIdx.x;
    int tile = blockIdx.x;            // 0..6
    int wave = t >> 5, lane = t & 31;
    int kg = lane >> 4, mn = lane & 15;

    // zero the weight rows whose column is out of range (never async-written)
    for (int idx = t; idx < 16 * 260; idx += 256) {
        int row = idx / 260;
        if (((tile << 4) + row) >= 100) (&w1s[0][0])[idx] = 0.0f;
    }

    v8f c = {};
    for (int ch = 0; ch < 32; ++ch) {
        int kc = ch << 8;
        for (int i = 0; i < 4; ++i) {
            int idx = (i << 8) + t;       // 0..1023 16-byte units
            int row = idx >> 6;           // 0..15
            int v4  = (idx & 63) << 2;    // float column 0..252 step 4
            async_copy_b128(lds_off_u32(&zs[row][v4]), z + (row << 13) + kc + v4);
            int col = (tile << 4) + row;
            if (col < 100)
                async_copy_b128(lds_off_u32(&w1s[row][v4]),
                                w1 + (size_t)col * 8192 + kc + v4);
        }
        async_wait0();
        __syncthreads();
        for (int s = 0; s < 8; ++s) {
            int kbl = (wave << 5) + (s << 2) + (kg << 1);
            v2f a, bm;
            a.x  = zs[mn][kbl];  a.y  = zs[mn][kbl + 1];
            bm.x = w1s[mn][kbl]; bm.y = w1s[mn][kbl + 1];
            c = __builtin_amdgcn_wmma_f32_16x16x4_f32(false, a, false, bm, (short)0, c, false, false);
        }
        __syncthreads();
    }
    for (int r = 0; r < 8; ++r)
        red[wave][((r + (kg << 3)) << 4) + mn] = c[r];
    __syncthreads();
    int m = t >> 4, nn = t & 15;
    float s = 0.0f;
    for (int v = 0; v < 8; ++v) s += red[v][t];
    int col = (tile << 4) + nn;
    if (col < 100) zf[m * 100 + col] = s + b1[col];
}

// ---------------------------------------------------------------------------
// K6: head: BN3(batch stats over 16 rows) + ReLU + fc2 (WMMA, 8 col tiles)
// single block, 256 threads.
// ---------------------------------------------------------------------------
__global__ void head_wmma_kernel(const float* __restrict__ zf,
                                 const float* __restrict__ g3, const float* __restrict__ b3,
                                 const float* __restrict__ w2, const float* __restrict__ b2,
                                 float* __restrict__ out) {
    __shared__ float as[16][104];
    __shared__ float s3[100], t3[100];
    int t = threadIdx.x;
    for (int idx = t; idx < 1600; idx += 256) {
        int row = idx / 100, col = idx - row * 100;
        as[row][col] = zf[idx];
    }
    __syncthreads();
    if (t < 100) {
        float sm = 0.0f, sq = 0.0f;
        for (int r = 0; r < 16; ++r) { float v = as[r][t]; sm += v; sq += v * v; }
        float m = sm * 0.0625f;
        float var = sq * 0.0625f - m * m;
        float sc = g3[t] * rsqrtf(var + BN_EPS);
        s3[t] = sc; t3[t] = b3[t] - m * sc;
    }
    __syncthreads();
    for (int idx = t; idx < 1600; idx += 256) {
        int row = idx / 100, col = idx - row * 100;
        float v = fmaf(as[row][col], s3[col], t3[col]);
        as[row][col] = v > 0.0f ? v : 0.0f;
    }
    __syncthreads();

    int wave = t >> 5, lane = t & 31;     // wave = output col tile (128/16 = 8)
    int kg = lane >> 4, mn = lane & 15;
    int col = (wave << 4) + mn;
    v8f c = {};
    for (int s = 0; s < 25; ++s) {        // K = 100 = 25 * 4
        int k0 = (s << 2) + (kg << 1);
        v2f a, bm;
        a.x  = as[mn][k0];         a.y  = as[mn][k0 + 1];
        bm.x = w2[col * 100 + k0]; bm.y = w2[col * 100 + k0 + 1];
        c = __builtin_amdgcn_wmma_f32_16x16x4_f32(false, a, false, bm, (short)0, c, false, false);
    }
    for (int r = 0; r < 8; ++r) {
        int m = r + (kg << 3);            // batch row
        out[(m << 7) + col] = c[r] + b2[col];
    }
}

// ---------------------------------------------------------------------------
extern "C" void kernel_launch(void* const* d_in, const int* in_sizes, int n_in,
                              void* d_out, int out_size, void* d_ws, size_t ws_size,
                              hipStream_t stream) {
    (void)in_sizes; (void)n_in; (void)out_size; (void)ws_size;
    const float* x       = (const float*)d_in[0];
    const float* bmat    = (const float*)d_in[1];
    const float* conv1_w = (const float*)d_in[2];
    const float* conv1_b = (const float*)d_in[3];
    const float* bn1_g   = (const float*)d_in[4];
    const float* bn1_b   = (const float*)d_in[5];
    const float* conv2_w = (const float*)d_in[6];
    const float* conv2_b = (const float*)d_in[7];
    const float* bn2_g   = (const float*)d_in[8];
    const float* bn2_b   = (const float*)d_in[9];
    const float* w1      = (const float*)d_in[10];
    const float* b1      = (const float*)d_in[11];
    const float* bn3_g   = (const float*)d_in[12];
    const float* bn3_b   = (const float*)d_in[13];
    const float* w2      = (const float*)d_in[14];
    const float* b2      = (const float*)d_in[15];

    float* ws = (float*)d_ws;
    float* h1    = ws;                      // 16*32*1024
    float* h2    = h1 + 16 * 32 * 1024;     // 16*64*1024
    float* z     = h2 + 16 * 64 * 1024;     // 16*8192
    float* zf    = z  + 16 * 8192;          // 1600 (pad 2048)
    float* stats = zf + 2048;               // 192 (pad 256)
    float* scl   = stats + 256;             // 192 (pad 256)
    float* s1sum = stats,       *s1sq = stats + 32;
    float* s2sum = stats + 64,  *s2sq = stats + 128;
    float* sc1 = scl,        *sh1 = scl + 32;
    float* sc2 = scl + 64,   *sh2 = scl + 128;

    zero_stats_kernel<<<1, 256, 0, stream>>>(stats);
    conv1_kernel<<<64, 256, 0, stream>>>(x, conv1_w, conv1_b, h1, s1sum, s1sq);
    bn_finalize_kernel<<<1, 64, 0, stream>>>(32, 1.0f / 16384.0f, s1sum, s1sq,
                                             bn1_g, bn1_b, sc1, sh1);
    conv2_wmma_kernel<<<512, 256, 0, stream>>>(h1, conv2_w, conv2_b, sc1, sh1,
                                               h2, s2sum, s2sq);
    bn_finalize_kernel<<<1, 64, 0, stream>>>(64, 1.0f / 16384.0f, s2sum, s2sq,
                                             bn2_g, bn2_b, sc2, sh2);
    pool_kernel<<<64, 256, 0, stream>>>(bmat, h2, sc2, sh2, z);
    fc1_wmma_kernel<<<7, 256, 0, stream>>>(z, w1, b1, zf);
    head_wmma_kernel<<<1, 256, 0, stream>>>(zf, bn3_g, bn3_b, w2, b2, (float*)d_out);
}